// ResGCN_33938831573082
// MI455X (gfx1250) — compile-verified
//
#include <hip/hip_runtime.h>

#define N_NODES 50000
#define N_EDGES 600000
#define D 128

typedef __attribute__((ext_vector_type(2))) float v2f;
typedef __attribute__((ext_vector_type(4))) float v4f;
typedef __attribute__((ext_vector_type(8))) float v8f;

// ---------------------------------------------------------------------------
// 1) zero the accumulation target (d_out) and the degree array
// ---------------------------------------------------------------------------
__global__ void __launch_bounds__(256) zero_kernel(float* __restrict__ out,
                                                   float* __restrict__ deg) {
    int i = blockIdx.x * blockDim.x + threadIdx.x;
    if (i < N_NODES * D) out[i] = 0.0f;
    if (i < N_NODES) deg[i] = 0.0f;
}

// ---------------------------------------------------------------------------
// 2) degree of each destination node (self-loops folded in).
//    f32 atomic counting is exact up to 2^24 >> max degree.
// ---------------------------------------------------------------------------
__global__ void __launch_bounds__(256) degree_kernel(const long long* __restrict__ ei,
                                                     float* __restrict__ deg) {
    int i = blockIdx.x * blockDim.x + threadIdx.x;
    if (i >= N_EDGES + N_NODES) return;
    int dst = (i < N_EDGES) ? (int)ei[N_EDGES + i] : (i - N_EDGES);
    atomicAdd(&deg[dst], 1.0f);
}

// ---------------------------------------------------------------------------
// 3) deg -> deg^{-1/2} in place
// ---------------------------------------------------------------------------
__global__ void __launch_bounds__(256) rsqrt_kernel(float* __restrict__ deg) {
    int i = blockIdx.x * blockDim.x + threadIdx.x;
    if (i >= N_NODES) return;
    float d = deg[i];
    deg[i] = (d > 0.0f) ? rsqrtf(d) : 0.0f;
}

// ---------------------------------------------------------------------------
// 4) xw = x @ W with fp32 WMMA (V_WMMA_F32_16X16X4_F32).
//    One wave32 owns a 16-row x 128-col output strip:
//      8 accumulator tiles (v8f each), K-loop of 32 steps of K=4.
//    N_NODES = 3125 * 16 exactly, D = 8 * 16 exactly -> no tails.
//    Fragment layouts per ISA 7.12.2:
//      A 16x4 : lanes 0-15 hold M=lane {K0,K1}; lanes 16-31 hold M=lane-16 {K2,K3}
//      B 4x16 : VGPR0 = rows K0(l0-15)/K2(l16-31); VGPR1 = rows K1/K3
//      C 16x16: VGPR v = row v (lanes 0-15) / row v+8 (lanes 16-31)
// ---------------------------------------------------------------------------
__global__ void __launch_bounds__(256) gemm_xw_kernel(const float* __restrict__ x,
                                                      const float* __restrict__ W,
                                                      float* __restrict__ xw) {
    const int lane   = threadIdx.x & 31;
    const int waveId = threadIdx.x >> 5;
    const int rowTile = blockIdx.x * (blockDim.x >> 5) + waveId;   // 0..3124
    if (rowTile >= N_NODES / 16) return;   // wave-uniform: EXEC all-ones inside

    const int row0 = rowTile * 16;
    const int m    = lane & 15;            // M index within tile
    const int koff = (lane >> 4) << 1;     // 0 for lanes 0-15, 2 for lanes 16-31

    v8f acc[8] = {};                       // 8 col-tiles of 16 -> 128 cols

    for (int k0 = 0; k0 < D; k0 += 4) {
        // A fragment: two consecutive K values for this lane's row
        v2f a = *(const v2f*)(x + (size_t)(row0 + m) * D + k0 + koff);

#pragma unroll
        for (int t = 0; t < 8; ++t) {
            const int c = t * 16 + m;      // N index
            v2f bfrag;
            bfrag.x = W[(size_t)(k0 + koff) * D + c];
            bfrag.y = W[(size_t)(k0 + koff + 1) * D + c];
            acc[t] = __builtin_amdgcn_wmma_f32_16x16x4_f32(
                /*neg_a=*/false, a, /*neg_b=*/false, bfrag,
                /*c_mod=*/(short)0, acc[t], /*reuse_a=*/false, /*reuse_b=*/false);
        }
    }

    // store C tiles
    const int rbase = row0 + ((lane < 16) ? 0 : 8);
#pragma unroll
    for (int t = 0; t < 8; ++t) {
        const int col = t * 16 + m;
#pragma unroll
        for (int v = 0; v < 8; ++v) {
            xw[(size_t)(rbase + v) * D + col] = acc[t][v];
        }
    }
}

// ---------------------------------------------------------------------------
// 5) edge scatter: one wave32 per (edge or self-loop), lane handles 4 floats.
//    agg[dst] += norm * xw[src]  via global_atomic_add_f32 (L2-resident).
// ---------------------------------------------------------------------------
__global__ void __launch_bounds__(256) scatter_kernel(const long long* __restrict__ ei,
                                                      const float* __restrict__ xw,
                                                      const float* __restrict__ dis,
                                                      float* __restrict__ out) {
    const int lane = threadIdx.x & 31;
    const long long i = (long long)blockIdx.x * (blockDim.x >> 5) + (threadIdx.x >> 5);
    if (i >= (long long)(N_EDGES + N_NODES)) return;

    int src, dst;
    if (i < N_EDGES) {
        src = (int)ei[i];
        dst = (int)ei[N_EDGES + i];
    } else {
        src = dst = (int)(i - N_EDGES);    // self-loop
    }
    const float norm = dis[src] * dis[dst];

    v4f v = ((const v4f*)(xw + (size_t)src * D))[lane];
    float* o = out + (size_t)dst * D + lane * 4;
    atomicAdd(o + 0, v.x * norm);
    atomicAdd(o + 1, v.y * norm);
    atomicAdd(o + 2, v.z * norm);
    atomicAdd(o + 3, v.w * norm);
}

// ---------------------------------------------------------------------------
// 6) out = out(agg) + x + b   (float4 vectorized)
// ---------------------------------------------------------------------------
__global__ void __launch_bounds__(256) finalize_kernel(const float* __restrict__ x,
                                                       const float* __restrict__ b,
                                                       float* __restrict__ out) {
    int i = blockIdx.x * blockDim.x + threadIdx.x;   // over N*D/4
    if (i >= N_NODES * D / 4) return;
    v4f o  = ((v4f*)out)[i];
    v4f xv = ((const v4f*)x)[i];
    v4f bv = ((const v4f*)b)[i & (D / 4 - 1)];
    o = o + xv + bv;
    ((v4f*)out)[i] = o;
}

// ---------------------------------------------------------------------------
extern "C" void kernel_launch(void* const* d_in, const int* in_sizes, int n_in,
                              void* d_out, int out_size, void* d_ws, size_t ws_size,
                              hipStream_t stream) {
    const float*     x  = (const float*)d_in[0];
    const long long* ei = (const long long*)d_in[1];   // int64 [2, E]
    const float*     W  = (const float*)d_in[2];
    const float*     b  = (const float*)d_in[3];
    float* out = (float*)d_out;

    // workspace: xw [N*D floats] + deg/dis [N floats]  (~25.8 MB)
    float* xw  = (float*)d_ws;
    float* deg = xw + (size_t)N_NODES * D;

    const int B = 256;

    // 1) zero accumulators
    zero_kernel<<<(N_NODES * D + B - 1) / B, B, 0, stream>>>(out, deg);

    // 2) degrees (edges + self-loops)
    degree_kernel<<<(N_EDGES + N_NODES + B - 1) / B, B, 0, stream>>>(ei, deg);

    // 3) deg^{-1/2}
    rsqrt_kernel<<<(N_NODES + B - 1) / B, B, 0, stream>>>(deg);

    // 4) xw = x @ W  (fp32 WMMA; 3125 row tiles, 8 waves/block)
    const int tiles = N_NODES / 16;                 // 3125
    const int wavesPerBlock = B / 32;               // 8
    gemm_xw_kernel<<<(tiles + wavesPerBlock - 1) / wavesPerBlock, B, 0, stream>>>(x, W, xw);

    // 5) scatter messages (one wave per edge/self-loop)
    const long long work = N_EDGES + N_NODES;
    scatter_kernel<<<(unsigned)((work + wavesPerBlock - 1) / wavesPerBlock), B, 0, stream>>>(
        ei, xw, deg, out);

    // 6) residual + bias
    finalize_kernel<<<(N_NODES * D / 4 + B - 1) / B, B, 0, stream>>>(x, b, out);
}